// QuantumLayer_53927609368584
// MI455X (gfx1250) — compile-verified
//
#include <hip/hip_runtime.h>
#include <math.h>

// Quantum patch layer, analytically collapsed:
//   out[b*784 + w*196 + 14*i + j] = - prod_{u<=w} cos(pi * x_u + theta_u)
// where (x0,x1,x2,x3) = (im[2i,2j], im[2i,2j+1], im[2i+1,2j], im[2i+1,2j+1]).
//
// Memory-bound kernel (~51 MB total traffic -> ~2.2 us at 23.3 TB/s).
// Uses the CDNA5 async global->LDS path (ASYNCcnt) to stage patch pixels.

#define PI_F 3.14159265358979323846f

__global__ __launch_bounds__(256) void quantum_patch_kernel(
    const float* __restrict__ images,  // [B,28,28]
    const float* __restrict__ theta,   // [4]
    float* __restrict__ out,           // [B,14,14,4] in the TF-scrambled layout
    int n_patches)                     // B*196
{
    __shared__ float tile[256 * 4];    // 4 pixels per thread, 4 KB per block

    const int tid = threadIdx.x;
    const unsigned pid = blockIdx.x * 256u + (unsigned)tid;
    if (pid >= (unsigned)n_patches) return;

    const unsigned b = pid / 196u;
    const unsigned p = pid - b * 196u;   // 14*i + j
    const unsigned i = p / 14u;
    const unsigned j = p - i * 14u;

    // Byte offsets of the two 8-byte row segments of this 2x2 patch.
    const unsigned g0 = (b * 784u + (2u * i) * 28u + 2u * j) * 4u; // row 2i
    const unsigned g1 = g0 + 28u * 4u;                             // row 2i+1

    // Raw LDS byte offsets for this lane's 16-byte slot (low 32 bits of the
    // generic address == LDS offset on gfx1250 flat addressing).
    const unsigned lds0 = (unsigned)(size_t)(&tile[tid * 4]);
    const unsigned lds1 = lds0 + 8u;

    // CDNA5 async tensor/LDS data path: per-lane gather of 8B straight into
    // LDS, tracked by ASYNCcnt. Each wave only consumes its own lanes' data,
    // so s_wait_asynccnt 0 (no barrier) is sufficient before the ds reads.
    asm volatile("global_load_async_to_lds_b64 %0, %1, %2"
                 :: "v"(lds0), "v"(g0), "s"(images) : "memory");
    asm volatile("global_load_async_to_lds_b64 %0, %1, %2"
                 :: "v"(lds1), "v"(g1), "s"(images) : "memory");
    asm volatile("s_wait_asynccnt 0" ::: "memory");

    const float x0 = tile[tid * 4 + 0];
    const float x1 = tile[tid * 4 + 1];
    const float x2 = tile[tid * 4 + 2];
    const float x3 = tile[tid * 4 + 3];

    // theta is uniform -> scalar loads
    const float t0 = theta[0];
    const float t1 = theta[1];
    const float t2 = theta[2];
    const float t3 = theta[3];

    // z_w = prod_{u<=w} cos(pi*x_u + theta_u); v_cos_f32 via __cosf.
    const float z0 = __cosf(fmaf(PI_F, x0, t0));
    const float z1 = z0 * __cosf(fmaf(PI_F, x1, t1));
    const float z2 = z1 * __cosf(fmaf(PI_F, x2, t2));
    const float z3 = z2 * __cosf(fmaf(PI_F, x3, t3));

    // Faithful output layout: flat-in-image index f = w*196 + p, negated.
    float* o = out + (size_t)b * 784u + p;
    o[0]   = -z0;
    o[196] = -z1;
    o[392] = -z2;
    o[588] = -z3;
}

extern "C" void kernel_launch(void* const* d_in, const int* in_sizes, int n_in,
                              void* d_out, int out_size, void* d_ws, size_t ws_size,
                              hipStream_t stream) {
    (void)n_in; (void)out_size; (void)d_ws; (void)ws_size;
    const float* images = (const float*)d_in[0]; // [B,28,28] fp32
    const float* theta  = (const float*)d_in[1]; // [4] fp32
    float* out = (float*)d_out;

    const int n_patches = in_sizes[0] / 4;       // (B*28*28)/4 = B*196
    const int blocks = (n_patches + 255) / 256;

    quantum_patch_kernel<<<blocks, 256, 0, stream>>>(images, theta, out, n_patches);
}